// GATModel_81123342287577
// MI455X (gfx1250) — compile-verified
//
#include <hip/hip_runtime.h>
#include <hip/hip_bf16.h>
#include <stdint.h>

#define NEG_SLOPE 0.2f

typedef float v2f __attribute__((ext_vector_type(2)));
typedef float v8f __attribute__((ext_vector_type(8)));

// ---------- ordered-int float max helpers (exact segment-max via int atomics) ----------
__device__ __forceinline__ int f2oi(float f) {
    int i = __float_as_int(f);
    return (i >= 0) ? i : (i ^ 0x7FFFFFFF);
}
__device__ __forceinline__ float oi2f(int i) {
    int j = (i >= 0) ? i : (i ^ 0x7FFFFFFF);
    return __int_as_float(j);
}

// =====================================================================================
// GEMM via V_WMMA_F32_16X16X4_F32:  H[nrows, NOUT] = A[nrows, K] @ B[K, NOUT]
// One wave computes one 16-row tile. B staged in LDS padded to 16 columns.
// A-operand layout (32-bit A 16x4): lane 0-15 -> M=lane, K={kb,kb+1};
//                                   lane 16-31 -> M=lane-16, K={kb+2,kb+3}.
// B-operand layout mirrors A with N in place of M.
// D layout: lane<16 holds rows 0..7 of col N=lane; lane>=16 holds rows 8..15 of col N=lane-16.
// =====================================================================================
template <int K, int NOUT>
__global__ void gemm_wmma_f32(const float* __restrict__ A,
                              const float* __restrict__ B,
                              float* __restrict__ H,
                              int nrows) {
    __shared__ float Bs[K * 16];
    const int tid = threadIdx.x;
    for (int i = tid; i < K * 16; i += blockDim.x) {
        const int k = i >> 4, c = i & 15;
        Bs[i] = (c < NOUT) ? B[k * NOUT + c] : 0.0f;
    }
    __syncthreads();

    const int lane  = tid & 31;
    const int wave  = tid >> 5;
    const int tile  = blockIdx.x * (blockDim.x >> 5) + wave;
    const int ntile = nrows >> 4;   // nrows divisible by 16
    if (tile >= ntile) return;      // wave-uniform: EXEC stays all-1 for WMMA

    const int row   = (tile << 4) + (lane & 15);
    const int khalf = (lane >> 4) << 1;      // 0 or 2
    const float* arow = A + (size_t)row * K + khalf;

    v8f acc = {};
#pragma unroll 8
    for (int kb = 0; kb < K; kb += 4) {
        v2f a = *(const v2f*)(arow + kb);    // global_load_b64, coalesced per half-wave
        v2f b;
        b.x = Bs[(kb + khalf) * 16 + (lane & 15)];
        b.y = Bs[(kb + khalf + 1) * 16 + (lane & 15)];
        acc = __builtin_amdgcn_wmma_f32_16x16x4_f32(
            /*neg_a=*/false, a, /*neg_b=*/false, b,
            /*c_mod=*/(short)0, acc, /*reuse_a=*/false, /*reuse_b=*/false);
    }

    const int n = lane & 15;
    if (n < NOUT) {
        const int rbase = (tile << 4) + ((lane >> 4) << 3);
#pragma unroll
        for (int j = 0; j < 8; ++j)
            H[(size_t)(rbase + j) * NOUT + n] = acc[j];
    }
}

// ---------- per-node attention scores: s = H[i,:] . a ----------
template <int D>
__global__ void scores_kernel(const float* __restrict__ H,
                              const float* __restrict__ a_src,
                              const float* __restrict__ a_dst,
                              float* __restrict__ s_src,
                              float* __restrict__ s_dst, int n) {
    const int i = blockIdx.x * blockDim.x + threadIdx.x;
    if (i >= n) return;
    const float* h = H + (size_t)i * D;
    float ss = 0.f, sd = 0.f;
#pragma unroll
    for (int c = 0; c < D; ++c) { float v = h[c]; ss += v * a_src[c]; sd += v * a_dst[c]; }
    s_src[i] = ss;
    s_dst[i] = sd;
}

// ---------- init emax = -inf (ordered INT_MIN), denom = 0, agg = 0 ----------
__global__ void init_kernel(int* __restrict__ emax, float* __restrict__ denom,
                            float* __restrict__ agg, int n, int aggElems) {
    const int i = blockIdx.x * blockDim.x + threadIdx.x;
    if (i < n) { emax[i] = INT32_MIN; denom[i] = 0.f; }
    for (int j = i; j < aggElems; j += gridDim.x * blockDim.x) agg[j] = 0.f;
}

// ---------- edge pass 1: logits + segment max ----------
__global__ void edge_logits(const int64_t* __restrict__ src, const int64_t* __restrict__ dst,
                            int E, int N,
                            const float* __restrict__ s_src, const float* __restrict__ s_dst,
                            float* __restrict__ ew, int* __restrict__ emax) {
    const int i = blockIdx.x * blockDim.x + threadIdx.x;
    const int tot = E + N;
    if (i >= tot) return;
    int s, d;
    if (i < E) { s = (int)src[i]; d = (int)dst[i]; } else { s = d = i - E; }
    float v = s_src[s] + s_dst[d];
    v = (v > 0.f) ? v : NEG_SLOPE * v;
    ew[i] = v;
    atomicMax(&emax[d], f2oi(v));
}

// ---------- edge pass 2: w = exp(e - emax[d]); denom[d] += w ----------
__global__ void edge_exp(const int64_t* __restrict__ dst, int E, int N,
                         float* __restrict__ ew, const int* __restrict__ emax,
                         float* __restrict__ denom) {
    const int i = blockIdx.x * blockDim.x + threadIdx.x;
    const int tot = E + N;
    if (i >= tot) return;
    const int d = (i < E) ? (int)dst[i] : (i - E);
    const float w = __expf(ew[i] - oi2f(emax[d]));
    ew[i] = w;
    atomicAdd(&denom[d], w);
}

// ---------- edge pass 3: agg[d,:] += alpha * h[s,:] ----------
template <int D>
__global__ void edge_aggregate(const int64_t* __restrict__ src, const int64_t* __restrict__ dst,
                               int E, int N,
                               const float* __restrict__ ew, const float* __restrict__ denom,
                               const float* __restrict__ H, float* __restrict__ agg) {
    const int i = blockIdx.x * blockDim.x + threadIdx.x;
    const int tot = E + N;
    if (i >= tot) return;
    int s, d;
    if (i < E) { s = (int)src[i]; d = (int)dst[i]; } else { s = d = i - E; }
    const float alpha = ew[i] / denom[d];
    const float* h = H + (size_t)s * D;
    float msg[D];
#pragma unroll
    for (int c = 0; c < D; ++c) msg[c] = alpha * h[c];
    float* o = agg + (size_t)d * D;
#pragma unroll
    for (int c = 0; c < D; ++c) atomicAdd(&o[c], msg[c]);
}

// ---------- bias + optional relu (out-of-place ok, here in-place on agg) ----------
template <int D, bool RELU>
__global__ void bias_act(float* __restrict__ z, const float* __restrict__ b, int n) {
    const int i = blockIdx.x * blockDim.x + threadIdx.x;
    if (i >= n * D) return;
    float v = z[i] + b[i % D];
    if (RELU) v = v > 0.f ? v : 0.f;
    z[i] = v;
}

// ---------- final FC: out[i,c] = z[i,:8] . Wfc[:,c] + bfc[c] ----------
__global__ void fc_kernel(const float* __restrict__ z, const float* __restrict__ Wfc,
                          const float* __restrict__ bfc, float* __restrict__ out, int n) {
    const int i = blockIdx.x * blockDim.x + threadIdx.x;
    if (i >= n * 16) return;
    const int node = i >> 4, c = i & 15;
    const float* zr = z + (size_t)node * 8;
    float acc = bfc[c];
#pragma unroll
    for (int j = 0; j < 8; ++j) acc += zr[j] * Wfc[j * 16 + c];
    out[i] = acc;
}

extern "C" void kernel_launch(void* const* d_in, const int* in_sizes, int n_in,
                              void* d_out, int out_size, void* d_ws, size_t ws_size,
                              hipStream_t stream) {
    const float*   x    = (const float*)  d_in[0];
    const int64_t* ei   = (const int64_t*)d_in[1];
    const float*   W1   = (const float*)  d_in[2];
    const float*   as1  = (const float*)  d_in[3];
    const float*   ad1  = (const float*)  d_in[4];
    const float*   b1   = (const float*)  d_in[5];
    const float*   W2   = (const float*)  d_in[6];
    const float*   as2  = (const float*)  d_in[7];
    const float*   ad2  = (const float*)  d_in[8];
    const float*   b2   = (const float*)  d_in[9];
    const float*   Wfc  = (const float*)  d_in[10];
    const float*   bfc  = (const float*)  d_in[11];
    float* out = (float*)d_out;

    const int N = in_sizes[0] / 256;
    const int E = in_sizes[1] / 2;
    const int ETOT = E + N;
    const int64_t* src = ei;
    const int64_t* dst = ei + E;

    // workspace carve-up (floats)
    float* ws   = (float*)d_ws;
    float* h1   = ws;              // N*16
    float* s_s  = h1  + (size_t)N * 16;  // N
    float* s_d  = s_s + N;               // N
    float* ew   = s_d + N;               // ETOT
    int*   emax = (int*)(ew + ETOT);     // N
    float* den  = (float*)(emax + N);    // N
    float* agg1 = den + N;               // N*16 (becomes z1 in-place)
    float* h2   = agg1 + (size_t)N * 16; // N*8
    float* agg2 = h2   + (size_t)N * 8;  // N*8 (becomes z2 in-place)

    const int TPB = 256;
    const int wavesPerBlk = TPB / 32;
    const int ntile = N / 16;
    const int gGemm = (ntile + wavesPerBlk - 1) / wavesPerBlk;
    const int gNode = (N + TPB - 1) / TPB;
    const int gEdge = (ETOT + TPB - 1) / TPB;

    // ---------------- layer 1 ----------------
    gemm_wmma_f32<256, 16><<<gGemm, TPB, 0, stream>>>(x, W1, h1, N);
    scores_kernel<16><<<gNode, TPB, 0, stream>>>(h1, as1, ad1, s_s, s_d, N);
    init_kernel<<<gNode, TPB, 0, stream>>>(emax, den, agg1, N, N * 16);
    edge_logits<<<gEdge, TPB, 0, stream>>>(src, dst, E, N, s_s, s_d, ew, emax);
    edge_exp<<<gEdge, TPB, 0, stream>>>(dst, E, N, ew, emax, den);
    edge_aggregate<16><<<gEdge, TPB, 0, stream>>>(src, dst, E, N, ew, den, h1, agg1);
    bias_act<16, true><<<(N * 16 + TPB - 1) / TPB, TPB, 0, stream>>>(agg1, b1, N);

    // ---------------- layer 2 ----------------
    gemm_wmma_f32<16, 8><<<gGemm, TPB, 0, stream>>>(agg1, W2, h2, N);
    scores_kernel<8><<<gNode, TPB, 0, stream>>>(h2, as2, ad2, s_s, s_d, N);
    init_kernel<<<gNode, TPB, 0, stream>>>(emax, den, agg2, N, N * 8);
    edge_logits<<<gEdge, TPB, 0, stream>>>(src, dst, E, N, s_s, s_d, ew, emax);
    edge_exp<<<gEdge, TPB, 0, stream>>>(dst, E, N, ew, emax, den);
    edge_aggregate<8><<<gEdge, TPB, 0, stream>>>(src, dst, E, N, ew, den, h2, agg2);
    bias_act<8, false><<<(N * 8 + TPB - 1) / TPB, TPB, 0, stream>>>(agg2, b2, N);

    // ---------------- classifier ----------------
    fc_kernel<<<(N * 16 + TPB - 1) / TPB, TPB, 0, stream>>>(agg2, Wfc, bfc, out, N);
}